// Cross_Attention_Block_3624952397825
// MI455X (gfx1250) — compile-verified
//
#include <hip/hip_runtime.h>
#include <hip/hip_bf16.h>

// ---------------------------------------------------------------------------
// Cross_Attention_Block — exact algebraic reduction.
// Output depends only on: (mean_k lidar) @ wv -> 3-layer leaky-ReLU MLP.
// (last query row of x is the zero pad row -> softmax is exactly uniform.)
// All GEMMs use V_WMMA_F32_16X16X4_F32 (M=16 == batch).
// ---------------------------------------------------------------------------

typedef __attribute__((ext_vector_type(2))) float v2f;
typedef __attribute__((ext_vector_type(8))) float v8f;

__device__ __forceinline__ v8f wmma4(v2f a, v2f b, v8f c) {
  // (neg_a, A, neg_b, B, c_mod, C, reuse_a, reuse_b)
  return __builtin_amdgcn_wmma_f32_16x16x4_f32(false, a, false, b,
                                               (short)0, c, false, false);
}

__device__ __forceinline__ float leaky(float v) {
  return v > 0.0f ? v : 0.01f * v;
}

// ---- Kernel 1: partial sums over lidar (B=16, N=4096, C=256) --------------
// grid (32 k-chunks, 16 batches), 256 threads. Each thread: float4 over 4
// channels, 32 strided rows. 512 blocks * 8 waves keeps HBM saturated.
__global__ __launch_bounds__(256) void k_partial(
    const float* __restrict__ lidar, float* __restrict__ partial) {
  const int kc   = blockIdx.x;        // 0..31  (128 points each)
  const int b    = blockIdx.y;        // 0..15
  const int tid  = threadIdx.x;       // 0..255
  const int c4   = (tid & 63) * 4;    // channel group
  const int ksub = tid >> 6;          // 0..3

  const float* base =
      lidar + ((size_t)b * 4096 + (size_t)kc * 128 + ksub) * 256 + c4;

  float4 acc = make_float4(0.f, 0.f, 0.f, 0.f);
#pragma unroll 8
  for (int i = 0; i < 32; ++i) {
    const float4 v = *reinterpret_cast<const float4*>(base + (size_t)i * 4 * 256);
    acc.x += v.x; acc.y += v.y; acc.z += v.z; acc.w += v.w;
  }

  __shared__ float4 red[256];
  red[tid] = acc;
  __syncthreads();
  if (ksub == 0) {
    float4 a0 = red[tid], a1 = red[tid + 64], a2 = red[tid + 128],
           a3 = red[tid + 192];
    float4 t;
    t.x = (a0.x + a1.x) + (a2.x + a3.x);
    t.y = (a0.y + a1.y) + (a2.y + a3.y);
    t.z = (a0.z + a1.z) + (a2.z + a3.z);
    t.w = (a0.w + a1.w) + (a2.w + a3.w);
    *reinterpret_cast<float4*>(partial + ((size_t)(b * 32 + kc)) * 256 + c4) = t;
  }
}

// ---- Kernel 2: finish mean = (1/4096) * sum of 32 partials ----------------
__global__ __launch_bounds__(256) void k_mean(
    const float* __restrict__ partial, float* __restrict__ mean) {
  const int b = blockIdx.x;
  const int c = threadIdx.x;
  float s = 0.0f;
#pragma unroll
  for (int kc = 0; kc < 32; ++kc)
    s += partial[((size_t)(b * 32 + kc)) * 256 + c];
  mean[b * 256 + c] = s * (1.0f / 4096.0f);
}

// ---- Kernel 3: mv(16x1024) = mean(16x256) @ wv(256x1024) via WMMA ---------
// 8 blocks x 8 waves; each wave owns one 16x16 N-tile, 64 K-steps of 4.
__global__ __launch_bounds__(256) void k_mv(
    const float* __restrict__ mean, const float* __restrict__ wv,
    float* __restrict__ mv) {
  const int wave = threadIdx.x >> 5;               // 0..7
  const int lane = threadIdx.x & 31;
  const int tile = blockIdx.x * 8 + wave;          // 0..63
  const int col0 = tile * 16;
  const int row  = lane & 15;                      // A: M-row, B: N-col
  const int kp   = lane >> 4;                      // K pair select
  const int n    = col0 + row;

  v8f acc;
#pragma unroll
  for (int r = 0; r < 8; ++r) acc[r] = 0.0f;

  for (int kk = 0; kk < 64; ++kk) {
    const int k0 = kk * 4 + 2 * kp;
    v2f a;  a[0] = mean[row * 256 + k0];
            a[1] = mean[row * 256 + k0 + 1];
    v2f bb; bb[0] = wv[(size_t)k0 * 1024 + n];
            bb[1] = wv[(size_t)(k0 + 1) * 1024 + n];
    acc = wmma4(a, bb, acc);
  }

  const int m0 = kp * 8;
#pragma unroll
  for (int r = 0; r < 8; ++r)
    mv[(size_t)(m0 + r) * 1024 + (col0 + row)] = acc[r];
}

// ---- Kernel 4: fused MLP, one workgroup of 16 waves -----------------------
// h1 = leaky(mv @ wo1 + b1)   (16x128, K=1024)  waves 0..7
// h2 = leaky(h1 @ wo2 + b2)   (16x128, K=128)   waves 0..7
// out = h2 @ wo3 + b3         (16x256, K=128)   waves 0..15
__global__ __launch_bounds__(512) void k_mlp(
    const float* __restrict__ mv,
    const float* __restrict__ wo1, const float* __restrict__ b1,
    const float* __restrict__ wo2, const float* __restrict__ b2,
    const float* __restrict__ wo3, const float* __restrict__ b3,
    float* __restrict__ out) {
  __shared__ float h1s[16 * 128];
  __shared__ float h2s[16 * 128];

  const int wave = threadIdx.x >> 5;   // 0..15
  const int lane = threadIdx.x & 31;
  const int row  = lane & 15;
  const int kp   = lane >> 4;
  const int m0   = kp * 8;

  // ---- layer 1: K = 1024 over mv ----
  if (wave < 8) {
    const int n = wave * 16 + row;
    const float bias = b1[n];
    v8f acc;
#pragma unroll
    for (int r = 0; r < 8; ++r) acc[r] = bias;
    for (int kk = 0; kk < 256; ++kk) {
      const int k0 = kk * 4 + 2 * kp;
      v2f a;  a[0] = mv[(size_t)row * 1024 + k0];
              a[1] = mv[(size_t)row * 1024 + k0 + 1];
      v2f bb; bb[0] = wo1[(size_t)k0 * 128 + n];
              bb[1] = wo1[(size_t)(k0 + 1) * 128 + n];
      acc = wmma4(a, bb, acc);
    }
#pragma unroll
    for (int r = 0; r < 8; ++r)
      h1s[(m0 + r) * 128 + n] = leaky(acc[r]);
  }
  __syncthreads();

  // ---- layer 2: K = 128 over h1 (LDS) ----
  if (wave < 8) {
    const int n = wave * 16 + row;
    const float bias = b2[n];
    v8f acc;
#pragma unroll
    for (int r = 0; r < 8; ++r) acc[r] = bias;
#pragma unroll 4
    for (int kk = 0; kk < 32; ++kk) {
      const int k0 = kk * 4 + 2 * kp;
      v2f a;  a[0] = h1s[row * 128 + k0];
              a[1] = h1s[row * 128 + k0 + 1];
      v2f bb; bb[0] = wo2[k0 * 128 + n];
              bb[1] = wo2[(k0 + 1) * 128 + n];
      acc = wmma4(a, bb, acc);
    }
#pragma unroll
    for (int r = 0; r < 8; ++r)
      h2s[(m0 + r) * 128 + n] = leaky(acc[r]);
  }
  __syncthreads();

  // ---- layer 3: K = 128 over h2 (LDS), N = 256 -> all 16 waves ----
  {
    const int n = wave * 16 + row;
    const float bias = b3[n];
    v8f acc;
#pragma unroll
    for (int r = 0; r < 8; ++r) acc[r] = bias;
#pragma unroll 4
    for (int kk = 0; kk < 32; ++kk) {
      const int k0 = kk * 4 + 2 * kp;
      v2f a;  a[0] = h2s[row * 128 + k0];
              a[1] = h2s[row * 128 + k0 + 1];
      v2f bb; bb[0] = wo3[(size_t)k0 * 256 + n];
              bb[1] = wo3[(size_t)(k0 + 1) * 256 + n];
      acc = wmma4(a, bb, acc);
    }
#pragma unroll
    for (int r = 0; r < 8; ++r)
      out[(size_t)(m0 + r) * 256 + n] = acc[r];
  }
}

// ---------------------------------------------------------------------------
extern "C" void kernel_launch(void* const* d_in, const int* in_sizes, int n_in,
                              void* d_out, int out_size, void* d_ws,
                              size_t ws_size, hipStream_t stream) {
  // setup_inputs order:
  // 0 feature, 1 lidar, 2 conv1_w, 3 conv2_w, 4 wq, 5 wk, 6 wv,
  // 7 wo1, 8 b1, 9 wo2, 10 b2, 11 wo3, 12 b3
  const float* lidar = (const float*)d_in[1];
  const float* wv    = (const float*)d_in[6];
  const float* wo1   = (const float*)d_in[7];
  const float* b1    = (const float*)d_in[8];
  const float* wo2   = (const float*)d_in[9];
  const float* b2    = (const float*)d_in[10];
  const float* wo3   = (const float*)d_in[11];
  const float* b3    = (const float*)d_in[12];

  float* ws      = (float*)d_ws;
  float* partial = ws;                   // 16*32*256 = 131072 floats
  float* mean    = ws + 131072;          // 16*256    =   4096 floats
  float* mv      = ws + 131072 + 4096;   // 16*1024   =  16384 floats

  k_partial<<<dim3(32, 16), dim3(256), 0, stream>>>(lidar, partial);
  k_mean  <<<dim3(16),      dim3(256), 0, stream>>>(partial, mean);
  k_mv    <<<dim3(8),       dim3(256), 0, stream>>>(mean, wv, mv);
  k_mlp   <<<dim3(1),       dim3(512), 0, stream>>>(mv, wo1, b1, wo2, b2,
                                                    wo3, b3, (float*)d_out);
}